// SoftConstrainedIPMModule_7249904796277
// MI455X (gfx1250) — compile-verified
//
#include <hip/hip_runtime.h>
#include <math.h>

typedef float v2f __attribute__((ext_vector_type(2)));
typedef float v8f __attribute__((ext_vector_type(8)));

#define BETA1f 0.5f
#define BETA2f 0.055f
#define N_OUTER 8
#define N_FEAS 12
#define N_RESID 12
#define NN 12
#define MM 11
#define BBATCH 16
#define WWID 512
#define PP (BBATCH*WWID)
#define BIGF 1e30f

__device__ inline float wave_sum(float v){
#pragma unroll
  for(int o=16;o;o>>=1) v += __shfl_xor(v,o,32);
  return v;
}
__device__ inline float wave_min(float v){
#pragma unroll
  for(int o=16;o;o>>=1) v = fminf(v,__shfl_xor(v,o,32));
  return v;
}
__device__ inline float wave_max(float v){
#pragma unroll
  for(int o=16;o;o>>=1) v = fmaxf(v,__shfl_xor(v,o,32));
  return v;
}
// SGPR broadcast of a known lane: v_readlane_b32, no LDS/DScnt traffic.
__device__ inline float readlane_f(float x, int lane){
  return __int_as_float(__builtin_amdgcn_readlane(__float_as_int(x), lane));
}
__device__ inline float hash_unif(unsigned x){
  x ^= x>>16; x *= 2654435761u; x ^= x>>13; x *= 0x85ebca6bu; x ^= x>>16;
  return (float)(x>>8) * (1.0f/16777216.0f);
}

// Phase 1: thread-per-problem, coalesced along W. Builds Hdiag, b, sorted S0,
// Lam0, alpha into 16-float padded SoA records in workspace.
__global__ void ipm_build_kernel(const float* __restrict__ cl_p,
                                 const float* __restrict__ Mu,
                                 const float* __restrict__ s2,
                                 const float* __restrict__ R,
                                 float* __restrict__ ws){
  const int p = blockIdx.x*blockDim.x + threadIdx.x;
  if (p >= PP) return;
  const int b = p >> 9, w = p & 511;
  const float cl = cl_p[0];
  const int base = b*NN*WWID + w;
  __builtin_prefetch(&Mu[base], 0, 0);
  __builtin_prefetch(&s2[base], 0, 0);
  __builtin_prefetch(&R [base], 0, 0);
  float mu[NN], sg[NN], rr[NN];
#pragma unroll
  for(int n=0;n<NN;++n){
    mu[n] = Mu[base + n*WWID];
    sg[n] = s2[base + n*WWID];
    rr[n] = R [base + n*WWID];
  }
  float* Hd = ws;
  float* bv = Hd + (size_t)PP*16;
  float* S0 = bv + (size_t)PP*16;
  float* L0 = S0 + (size_t)PP*16;
  float* al = L0 + (size_t)PP*16;
#pragma unroll
  for(int n=0;n<NN;++n){
    Hd[p*16+n] = cl/sg[n] + 2.0f + ((n<NN-1)?2.0f:0.0f);
    bv[p*16+n] = -cl*mu[n]/sg[n] - 2.0f*rr[n] + ((n<NN-1)? 2.0f*rr[n+1] : 0.0f);
  }
#pragma unroll
  for(int n=NN;n<16;++n){ Hd[p*16+n]=0.f; bv[p*16+n]=0.f; }
  // ascending sort of mu (static-index network -> stays in registers)
  float s[NN];
#pragma unroll
  for(int n=0;n<NN;++n) s[n]=mu[n];
#pragma unroll
  for(int i=0;i<NN;++i)
#pragma unroll
    for(int j=0;j<NN-1;++j){
      float a=s[j], c=s[j+1];
      s[j]=fminf(a,c); s[j+1]=fmaxf(a,c);
    }
#pragma unroll
  for(int n=0;n<NN;++n) S0[p*16+n]=s[n];
#pragma unroll
  for(int n=NN;n<16;++n) S0[p*16+n]=0.f;
#pragma unroll
  for(int m=0;m<16;++m)
    L0[p*16+m] = (m<MM)? hash_unif(0x9e370001u + (unsigned)(p*32+m)) : 0.f;
  al[p] = 10.0f + hash_unif(0xabcd1234u + (unsigned)p);
}

// Phase 2: one wave32 per problem. Schur complement reduces the 23x23 KKT
// system to a 12x12 tridiagonal system, padded to 16x16 and solved by blocked
// Gauss-Jordan. Each rank-4 step uses the uniform update C -= U @ C[blk,:]
// (U = W Pinv off-block, I - Pinv in-block), applied by one WMMA to the tile
// and one WMMA to its maintained transpose (which makes the column-panel
// gather a cheap row read). Pivot/rhs blocks broadcast via v_readlane.
__global__ void __launch_bounds__(256) ipm_solve_kernel(
                                const float* __restrict__ ws,
                                float* __restrict__ out){
  const int l = threadIdx.x & 31;
  const int p = blockIdx.x*(blockDim.x>>5) + (threadIdx.x>>5);
  if (p >= PP) return; // uniform per wave (grid sized exactly)
  const int b = p >> 9, w = p & 511;
  const float* Hdp = ws;
  const float* bvp = Hdp + (size_t)PP*16;
  const float* S0p = bvp + (size_t)PP*16;
  const float* L0p = S0p + (size_t)PP*16;
  const float* alp = L0p + (size_t)PP*16;
  const float Hd = (l<NN)? Hdp[p*16+l] : 0.f;
  const float bv = (l<NN)? bvp[p*16+l] : 0.f;
  float S0v = (l<NN)? S0p[p*16+l] : 0.f;   // carry S (lane n holds S[n])
  float L0v = (l<MM)? L0p[p*16+l] : 0.f;   // carry Lam
  const float alpha = alp[p];
  const int half = l>>4, col = l&15;
  const int lp1 = (l<31)? l+1 : l;
  const int lm1 = (l>0)? l-1 : 0;

#pragma unroll 1
  for(int outer=0; outer<N_OUTER; ++outer){
    // AS = A @ S  (first difference)
    float Sn  = __shfl(S0v, lp1, 32);
    float AS0 = (l<MM)? (S0v - Sn) : 0.f;
    float denom = wave_sum((l<MM)? AS0*L0v : 0.f);
    float t = -alpha*(float)MM/denom;
    float rcp_t = 1.0f/t;
    // residual at (S0c, Lam0c):  R1 = Q S + b + At Lam ; R2 = -Lam*AS - 1/t
    float Sp = __shfl(S0v, lm1, 32);
    float Lp = __shfl(L0v, lm1, 32);
    float R1 = (l<NN)? (Hd*S0v + bv + ((l>=1)? (-4.f*Sp - Lp):0.f)
                        + ((l<MM)? L0v:0.f)) : 0.f;
    float R2 = (l<MM)? (-L0v*AS0 - rcp_t) : 0.f;
    float RNorm = sqrtf(wave_sum(R1*R1 + R2*R2));
    // reduced tridiagonal system T x = rhs,  T = Q - At diag(Lam/AS) A
    float dcoef = (l<MM)? (L0v/AS0) : 0.f;
    float dprev = __shfl(dcoef, lm1, 32); dprev = (l>=1)? dprev : 0.f;
    float Tdiag = (l<NN)? (Hd - (((l<MM)?dcoef:0.f)+dprev)) : 0.f;
    float Tup   = (l<MM)? dcoef : 0.f;                    // T[n][n+1]
    float Tlo   = (l>=1 && l<NN)? (-4.f + dprev) : 0.f;   // T[n][n-1]
    float r2a   = (l<MM)? (R2/AS0) : 0.f;
    float r2ap  = __shfl(r2a, lm1, 32); r2ap = (l>=1)? r2ap : 0.f;
    float rhs   = (l<NN)? (-R1 - (((l<MM)?r2a:0.f) - r2ap)) : 0.f;

    // pack padded 16x16 tile: cc[v] = T[v+8*half][col]  (WMMA C/D layout)
    float cc[8];
#pragma unroll
    for(int v=0;v<8;++v){
      int r = v + 8*half;
      float dg = __shfl(Tdiag, r, 32);
      float up = __shfl(Tup,   r, 32);
      float lo = __shfl(Tlo,   r, 32);
      float val = 0.f;
      if(r < NN){
        if(col==r)        val = dg;
        else if(col==r+1) val = up;
        else if(col+1==r) val = lo;
      } else if (col==r)  val = 1.0f;   // identity padding rows 12..15
      cc[v]=val;
    }
    // transposed tile: tt[v] = T[col][v+8*half]
    float tt[8];
    {
      float dg_c = __shfl(Tdiag, col, 32);
      float up_c = __shfl(Tup,   col, 32);
      float lo_c = __shfl(Tlo,   col, 32);
#pragma unroll
      for(int v=0;v<8;++v){
        int r = v + 8*half;
        float val = 0.f;
        if(col < NN){
          if(r==col)        val = dg_c;
          else if(r==col+1) val = up_c;
          else if(r+1==col) val = lo_c;
        } else if (r==col)  val = 1.0f;
        tt[v]=val;
      }
    }
    float rhs_rep = __shfl(rhs, col, 32);   // replicated: lane holds rhs[col]

    // blocked Gauss-Jordan: 4 pivot blocks of 4, two WMMA rank-4 updates each
#pragma unroll
    for(int k=0;k<4;++k){
      const int hk = (k>=2) ? 1 : 0;
      // pivot block via v_readlane (constant lanes; no LDS traffic)
      float P[4][4], Pi[4][4];
#pragma unroll
      for(int i=0;i<4;++i)
#pragma unroll
        for(int j=0;j<4;++j){
          P[i][j]  = readlane_f(cc[(4*k+i)&7], (4*k+j)+16*hk);
          Pi[i][j] = (i==j)?1.f:0.f;
        }
      // 4x4 pivot inverse (redundant on all lanes)
#pragma unroll
      for(int cdx=0;cdx<4;++cdx){
        float pv = 1.0f/P[cdx][cdx];
#pragma unroll
        for(int j=0;j<4;++j){ P[cdx][j]*=pv; Pi[cdx][j]*=pv; }
#pragma unroll
        for(int i=0;i<4;++i){
          if(i==cdx) continue;
          float f=P[i][cdx];
#pragma unroll
          for(int j=0;j<4;++j){ P[i][j]-=f*P[cdx][j]; Pi[i][j]-=f*Pi[cdx][j]; }
        }
      }
      // row panel C[blk,:] and column panel W = C[:,blk] (via transpose tile):
      // single per-lane source (col + 16*hk) reused by all 8 gathers.
      const int gsrc = col + 16*hk;
      float Crow[4], Wr[4];
#pragma unroll
      for(int i=0;i<4;++i){
        Crow[i] = __shfl(cc[(4*k+i)&7], gsrc, 32);  // C[4k+i][col]
        Wr[i]   = __shfl(tt[(4*k+i)&7], gsrc, 32);  // C[col][4k+i]
      }
      // U[col][j]: W@Pinv off-block, I-Pinv in-block -> uniform GJ update
      const int mq = col - 4*k;
      const bool inblk = (mq>=0)&&(mq<4);
      float U0,U1,U2,U3;
#pragma unroll
      for(int j=0;j<4;++j){
        float un = Wr[0]*Pi[0][j]+Wr[1]*Pi[1][j]+Wr[2]*Pi[2][j]+Wr[3]*Pi[3][j];
        float u0 = ((j==0)?1.f:0.f)-Pi[0][j];
        float u1 = ((j==1)?1.f:0.f)-Pi[1][j];
        float u2 = ((j==2)?1.f:0.f)-Pi[2][j];
        float u3 = ((j==3)?1.f:0.f)-Pi[3][j];
        float ub = (mq==0)?u0:(mq==1)?u1:(mq==2)?u2:u3;
        float uj = inblk ? ub : un;
        if(j==0) U0=uj; else if(j==1) U1=uj; else if(j==2) U2=uj; else U3=uj;
      }
      const float UK0 = half? U2 : U0, UK1 = half? U3 : U1;
      const float CK0 = half? Crow[2] : Crow[0], CK1 = half? Crow[3] : Crow[1];
      // C -= U @ C[blk,:]   (handles elimination AND pivot-row scaling)
      {
        v2f Av; Av[0] = -UK0; Av[1] = -UK1;
        v2f Bv; Bv[0] =  CK0; Bv[1] =  CK1;
        v8f Cv;
#pragma unroll
        for(int v=0;v<8;++v) Cv[v]=cc[v];
        Cv = __builtin_amdgcn_wmma_f32_16x16x4_f32(false, Av, false, Bv,
                                                   (short)0, Cv, false, false);
#pragma unroll
        for(int v=0;v<8;++v) cc[v]=Cv[v];
      }
      // C^T -= C[blk,:]^T @ U^T  (keeps the transpose tile current)
      {
        v2f Av; Av[0] = -CK0; Av[1] = -CK1;
        v2f Bv; Bv[0] =  UK0; Bv[1] =  UK1;
        v8f Cv;
#pragma unroll
        for(int v=0;v<8;++v) Cv[v]=tt[v];
        Cv = __builtin_amdgcn_wmma_f32_16x16x4_f32(false, Av, false, Bv,
                                                   (short)0, Cv, false, false);
#pragma unroll
        for(int v=0;v<8;++v) tt[v]=Cv[v];
      }
      // rhs -= U @ rhs[blk]  (rhs[blk] -> Pinv@rhs[blk] falls out uniformly)
      float rb0 = readlane_f(rhs_rep, 4*k+0);
      float rb1 = readlane_f(rhs_rep, 4*k+1);
      float rb2 = readlane_f(rhs_rep, 4*k+2);
      float rb3 = readlane_f(rhs_rep, 4*k+3);
      rhs_rep -= U0*rb0 + U1*rb1 + U2*rb2 + U3*rb3;
    }
    float PDS  = (l<NN)? rhs_rep : 0.f;               // lane n: PD_S[n]
    float PDSn = __shfl(PDS, lp1, 32);
    float APDS = (l<MM)? (PDS - PDSn) : 0.f;
    float PDL  = (l<MM)? ((R2 - L0v*APDS)/AS0) : 0.f;
    // initial step
    float sm;
    if(l<MM){
      float npdl = (PDL<0.f)? PDL : -L0v;
      sm = -(L0v/npdl);
      if(sm != sm) sm = 1.0f;
    } else sm = BIGF;
    float step = 0.99f * wave_min(sm);
    float Sv = (l<NN)? (S0v + step*PDS) : 0.f;
    float ASv; { float x=__shfl(Sv, lp1, 32); ASv=(l<MM)?(Sv-x):0.f; }
    // feasibility backtracking
#pragma unroll 1
    for(int it=0; it<N_FEAS; ++it){
      float hp = wave_max((l<MM)? ASv : -BIGF);
      if(hp > 0.f) step *= BETA1f;      // uniform per wave
      Sv = (l<NN)? (S0v + step*PDS) : 0.f;
      float x=__shfl(Sv, lp1, 32); ASv=(l<MM)?(Sv-x):0.f;
    }
    float Lv = (l<MM)? (L0v + step*PDL) : 0.f;
    auto resnorm = [&](float S_, float L_, float AS_)->float{
      float Sp_=__shfl(S_, lm1, 32);
      float Lp_=__shfl(L_, lm1, 32);
      float r1=(l<NN)? (Hd*S_ + bv + ((l>=1)?(-4.f*Sp_-Lp_):0.f)
                        + ((l<MM)?L_:0.f)) : 0.f;
      float r2=(l<MM)? (-L_*AS_ - rcp_t) : 0.f;
      return sqrtf(wave_sum(r1*r1 + r2*r2));
    };
    float R2n = resnorm(Sv,Lv,ASv);
    // residual backtracking
#pragma unroll 1
    for(int it=0; it<N_RESID; ++it){
      if(R2n > (1.0f - BETA2f*step)*RNorm) step *= BETA1f;
      Sv=(l<NN)?(S0v+step*PDS):0.f;
      { float x=__shfl(Sv, lp1, 32); ASv=(l<MM)?(Sv-x):0.f; }
      Lv=(l<MM)?(L0v+step*PDL):0.f;
      R2n = resnorm(Sv,Lv,ASv);
    }
    S0v = Sv; L0v = Lv;
  }
  if(l<NN) out[b*(NN*WWID) + l*WWID + w] = S0v;
}

extern "C" void kernel_launch(void* const* d_in, const int* in_sizes, int n_in,
                              void* d_out, int out_size, void* d_ws, size_t ws_size,
                              hipStream_t stream){
  (void)in_sizes; (void)n_in; (void)out_size; (void)ws_size;
  const float* cl = (const float*)d_in[0];
  const float* Mu = (const float*)d_in[1];
  const float* s2 = (const float*)d_in[2];
  const float* R  = (const float*)d_in[3];
  float* out = (float*)d_out;
  float* ws  = (float*)d_ws;   // needs 4*PP*16 + PP floats (~2.1 MB)
  ipm_build_kernel<<<PP/256, 256, 0, stream>>>(cl, Mu, s2, R, ws);
  ipm_solve_kernel<<<PP/8, 256, 0, stream>>>(ws, out);
}